// LocalWLGNN_64630667870914
// MI455X (gfx1250) — compile-verified
//
#include <hip/hip_runtime.h>

typedef __attribute__((ext_vector_type(2))) float v2f;
typedef __attribute__((ext_vector_type(8))) float v8f;

#define GD 128        // feature dim
#define TPAD 130      // padded LDS row stride (floats) -> conflict-free transposed reads
#define MLP_WAVES 8   // waves per MLP workgroup

// out = (1 + eps) * x, float4-vectorized
__global__ void lwl_init_out(const float* __restrict__ x, const float* __restrict__ eps,
                             float* __restrict__ out, int n4) {
    int i = blockIdx.x * blockDim.x + threadIdx.x;
    if (i >= n4) return;
    float s = 1.0f + eps[0];
    float4 v = ((const float4*)x)[i];
    v.x *= s; v.y *= s; v.z *= s; v.w *= s;
    ((float4*)out)[i] = v;
}

// hp[dst] += x[src]; one wave (32 lanes) per edge, 4 floats per lane
__global__ void lwl_scatter(const float* __restrict__ x, float* __restrict__ hp,
                            const int* __restrict__ src, const int* __restrict__ dst, int E) {
    int t = blockIdx.x * blockDim.x + threadIdx.x;
    int e = t >> 5;
    int lane = t & 31;
    if (e >= E) return;
    int s = src[e];
    int d = dst[e];
    float4 v = ((const float4*)(x + (size_t)s * GD))[lane];
    float* o = hp + (size_t)d * GD + lane * 4;
    atomicAdd(o + 0, v.x);
    atomicAdd(o + 1, v.y);
    atomicAdd(o + 2, v.z);
    atomicAdd(o + 3, v.w);
}

// Repack W1,W2 (128x128 row-major) into WMMA B-fragment order:
// P[((tn*32 + ks)*32 + lane)] = { W[ks*4+kh][tn*16+m], W[ks*4+kh+1][tn*16+m] }
// with kh = (lane>>4)*2, m = lane&15  ->  one coalesced b64 per B-fragment.
__global__ void lwl_pack_w(const float* __restrict__ W1, const float* __restrict__ W2,
                           float* __restrict__ P1, float* __restrict__ P2) {
    int t = blockIdx.x * blockDim.x + threadIdx.x;   // 16384 threads total
    int lane = t & 31;
    int ks   = (t >> 5) & 31;
    int tn   = (t >> 10) & 7;
    int mat  = t >> 13;
    const float* W = mat ? W2 : W1;
    float*       P = mat ? P2 : P1;
    int kh = (lane >> 4) * 2;
    int m  = lane & 15;
    int k  = ks * 4 + kh;
    v2f v;
    v.x = W[(size_t)(k + 0) * GD + tn * 16 + m];
    v.y = W[(size_t)(k + 1) * GD + tn * 16 + m];
    ((v2f*)P)[(tn * 32 + ks) * 32 + lane] = v;
}

// out_tile += relu(hp_tile @ W1) @ W2 ; one wave per 16-row tile, f32 WMMA 16x16x4.
// A-fragments hoisted into registers (loaded once, reused across all 8 col tiles);
// B-fragments are single b64 loads from the pre-swizzled weight buffers.
__global__ void __launch_bounds__(MLP_WAVES * 32) lwl_mlp(
        const float* __restrict__ hp, const float* __restrict__ P1,
        const float* __restrict__ P2, float* __restrict__ out, int nrows) {
    __shared__ float tbuf[MLP_WAVES * 16 * TPAD];    // wave-private padded tiles
    const int wave = threadIdx.x >> 5;
    const int lane = threadIdx.x & 31;
    const int row0 = (blockIdx.x * MLP_WAVES + wave) * 16;
    if (row0 >= nrows) return;                       // wave-uniform exit, EXEC all-ones
    float* t = tbuf + wave * 16 * TPAD;
    const int m  = lane & 15;                        // M (A/C) or N (B/C) index
    const int kh = (lane >> 4) * 2;                  // K pair selected by half-wave
    const int rb = (lane >> 4) * 8;                  // C/D row base per half-wave
    const float* arow = hp + (size_t)(row0 + m) * GD;

    // ---- hoist stage-1 A-fragments: whole 16xD tile, 64 floats/lane ----
    v2f afr[32];
    #pragma unroll
    for (int ks = 0; ks < 32; ++ks)
        afr[ks] = *(const v2f*)(arow + ks * 4 + kh);

    // ---- stage 1: t = relu(hp_tile @ W1) ----
    const v2f* B1 = (const v2f*)P1;
    for (int tn = 0; tn < 8; ++tn) {
        v8f acc = {};
        const v2f* bp = B1 + (size_t)tn * 32 * 32 + lane;
        #pragma unroll
        for (int ks = 0; ks < 32; ++ks) {
            v2f b = bp[ks * 32];                     // one b64, immediate offset
            acc = __builtin_amdgcn_wmma_f32_16x16x4_f32(
                false, afr[ks], false, b, (short)0, acc, false, false);
        }
        #pragma unroll
        for (int g = 0; g < 8; ++g) {                // C: row rb+g, col tn*16+m
            float v = acc[g];
            t[(rb + g) * TPAD + tn * 16 + m] = v > 0.0f ? v : 0.0f;   // relu
        }
    }

    // ---- hoist stage-2 A-fragments of t from LDS (transposed read) ----
    v2f tfr[32];
    #pragma unroll
    for (int ks = 0; ks < 32; ++ks)
        tfr[ks] = *(const v2f*)(t + m * TPAD + ks * 4 + kh);

    // ---- stage 2: out_tile += t @ W2 ----
    const v2f* B2 = (const v2f*)P2;
    for (int tn = 0; tn < 8; ++tn) {
        v8f acc = {};
        const v2f* bp = B2 + (size_t)tn * 32 * 32 + lane;
        #pragma unroll
        for (int ks = 0; ks < 32; ++ks) {
            v2f b = bp[ks * 32];
            acc = __builtin_amdgcn_wmma_f32_16x16x4_f32(
                false, tfr[ks], false, b, (short)0, acc, false, false);
        }
        #pragma unroll
        for (int g = 0; g < 8; ++g) {
            size_t oi = (size_t)(row0 + rb + g) * GD + tn * 16 + m;
            out[oi] += acc[g];                       // each element owned by one lane
        }
    }
}

extern "C" void kernel_launch(void* const* d_in, const int* in_sizes, int n_in,
                              void* d_out, int out_size, void* d_ws, size_t ws_size,
                              hipStream_t stream) {
    const float* x0  = (const float*)d_in[0];
    const float* w1  = (const float*)d_in[1];
    const float* w2  = (const float*)d_in[2];
    const float* eps = (const float*)d_in[3];
    const int*   sidx = (const int*)d_in[4];  // gather source nodes per hop [H,E]
    const int*   didx = (const int*)d_in[5];  // scatter destination nodes per hop [H,E]
    float* out = (float*)d_out;

    const int N = 50000, H = 3, L = 2, E = 500000;
    const size_t ND = (size_t)N * GD;

    float* buf1 = (float*)d_ws;       // layer-0 accumulator / layer-1 input (25.6 MB)
    float* hp   = buf1 + ND;          // per-hop pre-MLP buffer (25.6 MB)
    float* P1   = hp + ND;            // swizzled W1 (64 KB)
    float* P2   = P1 + GD * GD;       // swizzled W2 (64 KB)

    const int initBlocks = (int)((ND / 4 + 255) / 256);
    const int scatBlocks = (int)(((size_t)E * 32 + 255) / 256);
    const int mlpBlocks  = (N / 16 + MLP_WAVES - 1) / MLP_WAVES;

    for (int l = 0; l < L; ++l) {
        const float* xin = (l == 0) ? x0 : buf1;
        float* oacc = (l == L - 1) ? out : buf1;
        lwl_init_out<<<initBlocks, 256, 0, stream>>>(xin, eps, oacc, (int)(ND / 4));
        for (int hop = 0; hop < H; ++hop) {
            hipMemcpyAsync(hp, xin, ND * sizeof(float), hipMemcpyDeviceToDevice, stream);
            lwl_scatter<<<scatBlocks, 256, 0, stream>>>(
                xin, hp, sidx + (size_t)hop * E, didx + (size_t)hop * E, E);
            const float* W1 = w1 + ((size_t)l * H + hop) * GD * GD;
            const float* W2 = w2 + ((size_t)l * H + hop) * GD * GD;
            lwl_pack_w<<<64, 256, 0, stream>>>(W1, W2, P1, P2);
            lwl_mlp<<<mlpBlocks, MLP_WAVES * 32, 0, stream>>>(hp, P1, P2, oacc, N);
        }
    }
}